// TSPEnergyFunction_62904091017756
// MI455X (gfx1250) — compile-verified
//
#include <hip/hip_runtime.h>

// TSP energy: out[b,i] = sum_j exp(-0.8|i-j|) * (1 - cos_sim(x1[b,i], x2[b,j]))
// Banded FP32-WMMA implementation for gfx1250 (MI455X).

typedef __attribute__((ext_vector_type(2))) float v2f;
typedef __attribute__((ext_vector_type(8))) float v8f;

#define B_      8
#define L_      2048
#define D_      1024
#define TILE    16
#define NT      (L_ / TILE)     // 128 tiles along the sequence
#define JT_HALF 3               // band half-width in tiles (weights < 1e-17 outside)
#define DECAYF  0.8f
#define EPSF    1e-8f

// ---------------------------------------------------------------------------
// Kernel 1: inverse row norms for both inputs.
// invn[row]          = 1/max(||x1_row||, eps)   for row in [0, 16384)
// invn[16384 + row]  = 1/max(||x2_row||, eps)
// One 256-thread block per row; each thread owns 4 contiguous floats.
// ---------------------------------------------------------------------------
__global__ void inv_norm_kernel(const float* __restrict__ x1,
                                const float* __restrict__ x2,
                                float* __restrict__ invn) {
    const int row = blockIdx.x;                       // 0 .. 2*B_*L_-1
    const int nrows = B_ * L_;
    const float* src = (row < nrows)
        ? (x1 + (size_t)row * D_)
        : (x2 + (size_t)(row - nrows) * D_);

    const int tid = threadIdx.x;                      // 256 threads
    float4 v = ((const float4*)src)[tid];             // 4 floats per thread
    float ss = v.x * v.x + v.y * v.y + v.z * v.z + v.w * v.w;

    // wave32 butterfly reduction
    #pragma unroll
    for (int off = 16; off > 0; off >>= 1)
        ss += __shfl_xor(ss, off, 32);

    __shared__ float red[8];
    const int wave = tid >> 5;
    const int lane = tid & 31;
    if (lane == 0) red[wave] = ss;
    __syncthreads();
    if (tid == 0) {
        float t = 0.0f;
        #pragma unroll
        for (int w = 0; w < 8; ++w) t += red[w];
        invn[row] = 1.0f / fmaxf(sqrtf(t), EPSF);
    }
}

// ---------------------------------------------------------------------------
// Kernel 2: banded similarity via V_WMMA_F32_16X16X4_F32.
// Grid: (NT, B_). One workgroup per (i-tile, batch); waves 0..6 each own
// one j-tile in the band jt = it-3 .. it+3 (clamped). Each wave computes a
// full 16x16 raw-dot tile over K=1024 (256 WMMAs across 4 accumulators),
// rescales by the two inverse norms (bilinearity), applies the positional
// weight, and reduces into a shared 16-entry row accumulator.
//
// FP32 16x16x4 fragment layout (ISA 7.12.2): for lane L (h = L>>4, m = L&15)
//   A vgpr v holds A[M=m, K = 2h+v]  -> one float2 load per k-step
//   B vgpr v holds B[K = 2h+v, N=m]  -> x2 row j0+m, same addressing
//   D vgpr r holds D[M = r + 8h, N = m]
// ---------------------------------------------------------------------------
__global__ void __launch_bounds__(256, 2)
tsp_energy_wmma_kernel(const float* __restrict__ x1,
                       const float* __restrict__ x2,
                       const float* __restrict__ invn,
                       float* __restrict__ out) {
    const int it = blockIdx.x;            // i-tile index, 0..127
    const int b  = blockIdx.y;            // batch, 0..7
    const int i0 = it * TILE;

    const int tid  = threadIdx.x;
    const int wave = tid >> 5;
    const int lane = tid & 31;
    const int h    = lane >> 4;           // half-wave selector
    const int m    = lane & 15;           // row (A) / column (B) within tile

    __shared__ float acc[TILE];
    if (tid < TILE) acc[tid] = 0.0f;
    __syncthreads();

    const int jt = it - JT_HALF + wave;   // wave-uniform j-tile assignment
    if (wave < (2 * JT_HALF + 1) && jt >= 0 && jt < NT) {
        const int j0 = jt * TILE;

        // Per-lane fragment base pointers (float2 per k-step of 4).
        const float* arow = x1 + ((size_t)(b * L_ + i0 + m) * D_ + 2 * h);
        const float* brow = x2 + ((size_t)(b * L_ + j0 + m) * D_ + 2 * h);

        v8f c0 = {}, c1 = {}, c2 = {}, c3 = {};
        for (int kk = 0; kk < D_; kk += 16) {
            v2f a0 = *(const v2f*)(arow + kk);
            v2f b0 = *(const v2f*)(brow + kk);
            v2f a1 = *(const v2f*)(arow + kk + 4);
            v2f b1 = *(const v2f*)(brow + kk + 4);
            v2f a2 = *(const v2f*)(arow + kk + 8);
            v2f b2 = *(const v2f*)(brow + kk + 8);
            v2f a3 = *(const v2f*)(arow + kk + 12);
            v2f b3 = *(const v2f*)(brow + kk + 12);
            c0 = __builtin_amdgcn_wmma_f32_16x16x4_f32(false, a0, false, b0,
                                                       (short)0, c0, false, false);
            c1 = __builtin_amdgcn_wmma_f32_16x16x4_f32(false, a1, false, b1,
                                                       (short)0, c1, false, false);
            c2 = __builtin_amdgcn_wmma_f32_16x16x4_f32(false, a2, false, b2,
                                                       (short)0, c2, false, false);
            c3 = __builtin_amdgcn_wmma_f32_16x16x4_f32(false, a3, false, b3,
                                                       (short)0, c3, false, false);
        }
        v8f c = c0 + c1 + c2 + c3;        // 16x16 raw-dot tile

        const int   j   = j0 + m;                     // this lane's column
        const float wn2 = invn[B_ * L_ + b * L_ + j]; // 1/||x2_j||

        #pragma unroll
        for (int r = 0; r < 8; ++r) {
            const int   Mi  = r + 8 * h;
            const int   i   = i0 + Mi;
            const float sim = c[r] * invn[b * L_ + i] * wn2;
            const float pw  = __expf(-DECAYF * fabsf((float)(i - j)));
            float contrib   = pw * (1.0f - sim);
            // reduce over the 16 lanes of this half (xor masks stay in-half)
            contrib += __shfl_xor(contrib, 1, 32);
            contrib += __shfl_xor(contrib, 2, 32);
            contrib += __shfl_xor(contrib, 4, 32);
            contrib += __shfl_xor(contrib, 8, 32);
            if (m == 0) atomicAdd(&acc[Mi], contrib);  // ds_add_f32 across waves
        }
    }
    __syncthreads();
    if (tid < TILE) out[(size_t)b * L_ + i0 + tid] = acc[tid];
}

// ---------------------------------------------------------------------------
extern "C" void kernel_launch(void* const* d_in, const int* in_sizes, int n_in,
                              void* d_out, int out_size, void* d_ws, size_t ws_size,
                              hipStream_t stream) {
    const float* x1 = (const float*)d_in[0];
    const float* x2 = (const float*)d_in[1];
    float* out  = (float*)d_out;          // [8, 2048]
    float* invn = (float*)d_ws;           // 2 * 16384 floats = 128 KB scratch

    inv_norm_kernel<<<dim3(2 * B_ * L_), dim3(256), 0, stream>>>(x1, x2, invn);
    tsp_energy_wmma_kernel<<<dim3(NT, B_), dim3(256), 0, stream>>>(x1, x2, invn, out);
}